// CRFLayer_82901458747418
// MI455X (gfx1250) — compile-verified
//
#include <hip/hip_runtime.h>
#include <stdint.h>

// Viterbi decode (CRF): B=512, S=1024, T=48.
// Forward: wave-per-batch max-plus scan, transitions in VGPRs, scores in LDS
//          (read as float4 broadcasts), 4-way-split argmax chains to shorten
//          the per-step critical path, emissions double-buffered via async
//          global->LDS (ASYNCcnt).
// Backward: chunked pointer chase; 32 backpointer rows prefetched to LDS per
//          chunk via async-b128, resolved via LDS broadcast byte loads.

#define TAGS 48
#define BATCH 512
#define SEQ 1024
#define WPB 8   // waves (batches) per block
#define NEG_INF (-3.0e38f)

#define AS1 __attribute__((address_space(1)))
#define AS3 __attribute__((address_space(3)))

typedef int v4i __attribute__((__vector_size__(16)));

#if __has_builtin(__builtin_amdgcn_global_load_async_to_lds_b128)
#define HAVE_ASYNC 1
#else
#define HAVE_ASYNC 0
#endif

__device__ __forceinline__ void async_cp16(const void* g, void* l) {
#if HAVE_ASYNC
  __builtin_amdgcn_global_load_async_to_lds_b128(
      (AS1 v4i*)g, (AS3 v4i*)l, 0, 0);
#endif
}
__device__ __forceinline__ void wait_async0() {
#if HAVE_ASYNC
  asm volatile("s_wait_asynccnt 0" ::: "memory");
#endif
}
__device__ __forceinline__ void wait_async1() {
#if HAVE_ASYNC
  asm volatile("s_wait_asynccnt 1" ::: "memory");
#endif
}
__device__ __forceinline__ void wait_async3() {
#if HAVE_ASYNC
  asm volatile("s_wait_asynccnt 3" ::: "memory");
#endif
}

// argmax accumulate with strict '>' (keeps lowest index on ties, matching
// jnp.argmax first-max semantics when scanned in ascending index order).
__device__ __forceinline__ void amax_acc(float v, int i, float& best, int& idx) {
  if (v > best) { best = v; idx = i; }
}

// ---------------------------------------------------------------------------
// Forward pass: one wave32 per batch. Lane l owns tags l and l+32 (lanes>=16
// only own one tag). No cross-wave sync needed: LDS ops from one wave are
// processed in order.
// ---------------------------------------------------------------------------
__global__ __launch_bounds__(WPB * 32) void viterbi_forward(
    const float* __restrict__ em,      // [B, S, T]
    const float* __restrict__ trans,   // [T, T]
    const float* __restrict__ startT,  // [T]
    const float* __restrict__ endT,    // [T]
    uint8_t* __restrict__ bp,          // [S-1, B, T] backpointers (u8)
    int* __restrict__ finalTag)        // [B]
{
  __shared__ __align__(16) float sT[TAGS * TAGS];
  __shared__ __align__(16) float sEm[WPB][2][TAGS];
  __shared__ __align__(16) float sScore[WPB][TAGS];

  const int tid  = threadIdx.x;
  const int w    = tid >> 5;
  const int lane = tid & 31;
  const int b    = blockIdx.x * WPB + w;
  const bool two = (lane < 16);
  const int j0   = lane;
  const int j1   = two ? (lane + 32) : 0;  // clamped for inactive lanes

  for (int k = tid; k < TAGS * TAGS; k += WPB * 32) sT[k] = trans[k];

  const float* emB = em + (size_t)b * (SEQ * TAGS);

  // t = 0 init
  sScore[w][j0] = startT[j0] + emB[j0];
  if (two) sScore[w][lane + 32] = startT[lane + 32] + emB[lane + 32];

  __syncthreads();  // sT ready for all waves

  // Transition columns for this lane's tags -> registers (constant over time).
  float colA[TAGS], colB[TAGS];
#pragma unroll
  for (int i = 0; i < TAGS; ++i) {
    colA[i] = sT[i * TAGS + j0];
    colB[i] = sT[i * TAGS + j1];
  }

  // Stage emissions for t=1 (row = 192 B = 12 lanes x 16 B).
#if HAVE_ASYNC
  if (lane < 12) async_cp16(emB + TAGS + lane * 4, &sEm[w][1][lane * 4]);
#endif

  for (int t = 1; t < SEQ; ++t) {
    const int nt = (t + 1 < SEQ) ? (t + 1) : t;  // dummy refetch on last iter
    const int pf = (t + 8 < SEQ) ? (t + 8) : (SEQ - 1);
    __builtin_prefetch(emB + (size_t)pf * TAGS + lane, 0, 1);  // global_prefetch_b8
#if HAVE_ASYNC
    if (lane < 12)
      async_cp16(emB + (size_t)nt * TAGS + lane * 4, &sEm[w][(t + 1) & 1][lane * 4]);
    wait_async1();  // in-order completion => buffer for step t is ready
    const float e0 = sEm[w][t & 1][j0];
    const float e1 = sEm[w][t & 1][j1];
#else
    const float e0 = emB[(size_t)t * TAGS + j0];
    const float e1 = two ? emB[(size_t)t * TAGS + lane + 32] : 0.0f;
#endif
    // 4 independent argmax chains (blocked: chain c covers i in [12c, 12c+12))
    // to cut the serial compare/select depth ~4x; scores read as float4
    // broadcasts (ds_load_b128, conflict-free).
    float best0[4], best1[4];
    int idx0[4], idx1[4];
#pragma unroll
    for (int c = 0; c < 4; ++c) {
      best0[c] = NEG_INF; best1[c] = NEG_INF; idx0[c] = 0; idx1[c] = 0;
    }
#pragma unroll
    for (int q = 0; q < 12; ++q) {   // quad q covers i = 4q .. 4q+3
      const int c = q / 3;           // 3 quads per chain
      const float4 s4 = *(const float4*)&sScore[w][q * 4];
      const int i0 = q * 4;
      amax_acc(s4.x + colA[i0 + 0], i0 + 0, best0[c], idx0[c]);
      amax_acc(s4.y + colA[i0 + 1], i0 + 1, best0[c], idx0[c]);
      amax_acc(s4.z + colA[i0 + 2], i0 + 2, best0[c], idx0[c]);
      amax_acc(s4.w + colA[i0 + 3], i0 + 3, best0[c], idx0[c]);
      amax_acc(s4.x + colB[i0 + 0], i0 + 0, best1[c], idx1[c]);
      amax_acc(s4.y + colB[i0 + 1], i0 + 1, best1[c], idx1[c]);
      amax_acc(s4.z + colB[i0 + 2], i0 + 2, best1[c], idx1[c]);
      amax_acc(s4.w + colB[i0 + 3], i0 + 3, best1[c], idx1[c]);
    }
    // Merge ascending with strict '>' -> lowest index wins ties (blocked
    // partition keeps chain indices ordered).
    float b0 = best0[0], b1 = best1[0];
    int bi0 = idx0[0], bi1 = idx1[0];
#pragma unroll
    for (int c = 1; c < 4; ++c) {
      if (best0[c] > b0) { b0 = best0[c]; bi0 = idx0[c]; }
      if (best1[c] > b1) { b1 = best1[c]; bi1 = idx1[c]; }
    }

    const size_t row = ((size_t)(t - 1) * BATCH + b) * TAGS;
    bp[row + j0] = (uint8_t)bi0;
    sScore[w][j0] = b0 + e0;
    if (two) {
      bp[row + lane + 32] = (uint8_t)bi1;
      sScore[w][lane + 32] = b1 + e1;
    }
    __builtin_amdgcn_wave_barrier();  // keep per-step LDS RAW order
  }

  // Final argmax over (scores + end_transitions), lowest index wins ties.
  float v = sScore[w][j0] + endT[j0];
  int bi = j0;
  if (two) {
    const float v1 = sScore[w][lane + 32] + endT[lane + 32];
    if (v1 > v) { v = v1; bi = lane + 32; }
  }
#pragma unroll
  for (int off = 16; off >= 1; off >>= 1) {
    const float ov = __shfl_xor(v, off, 32);
    const int   oi = __shfl_xor(bi, off, 32);
    if (ov > v || (ov == v && oi < bi)) { v = ov; bi = oi; }
  }
  if (lane == 0) finalTag[b] = bi;
}

// ---------------------------------------------------------------------------
// Backtrace: one wave32 per batch. Prefetch 32 bp rows (48 B each) per chunk
// into LDS (3 async b128 per chunk, double-buffered), resolve via dependent
// LDS broadcast byte loads; lane l records the tag of step l for a coalesced
// 32-wide store. The bp buffer (25 MB) is L2-resident (192 MB L2).
// ---------------------------------------------------------------------------
__global__ __launch_bounds__(WPB * 32) void viterbi_backtrace(
    const uint8_t* __restrict__ bp,      // [S-1, B, T]
    const int* __restrict__ finalTag,    // [B]
    float* __restrict__ out)             // [B, S] tags as float
{
  __shared__ __align__(16) uint8_t rows[WPB][2][32 * TAGS];  // 24 KB

  const int tid  = threadIdx.x;
  const int w    = tid >> 5;
  const int lane = tid & 31;
  const int b    = blockIdx.x * WPB + w;

  int cur = finalTag[b];
  if (lane == 0) out[(size_t)b * SEQ + (SEQ - 1)] = (float)cur;

  // Prefetch first chunk (steps 1022..991) into buffer 0.
  {
    const int t = (SEQ - 2) - lane;  // always >= 0 here
    const uint8_t* src = bp + ((size_t)t * BATCH + b) * TAGS;
    uint8_t* dst = &rows[w][0][lane * TAGS];
#if HAVE_ASYNC
    async_cp16(src, dst);
    async_cp16(src + 16, dst + 16);
    async_cp16(src + 32, dst + 32);
#else
    const uint4* s4 = (const uint4*)src;
    uint4* d4 = (uint4*)dst;
    d4[0] = s4[0]; d4[1] = s4[1]; d4[2] = s4[2];
#endif
  }

  int buf = 0;
  for (int cbase = SEQ - 2; cbase >= 0; cbase -= 32) {
    const int nbase = cbase - 32;
    if (nbase >= 0) {
      const int t = nbase - lane;
      if (t >= 0) {  // partial EXEC ok; 3 async ops still issue per wave
        const uint8_t* src = bp + ((size_t)t * BATCH + b) * TAGS;
        uint8_t* dst = &rows[w][buf ^ 1][lane * TAGS];
#if HAVE_ASYNC
        async_cp16(src, dst);
        async_cp16(src + 16, dst + 16);
        async_cp16(src + 32, dst + 32);
#else
        const uint4* s4 = (const uint4*)src;
        uint4* d4 = (uint4*)dst;
        d4[0] = s4[0]; d4[1] = s4[1]; d4[2] = s4[2];
#endif
      }
      wait_async3();  // in-order: current chunk's 3 copies complete
    } else {
      wait_async0();
    }

    const int nrows = (cbase + 1 < 32) ? (cbase + 1) : 32;
    const uint8_t* rb = &rows[w][buf][0];
    int mytag = 0;
    for (int l = 0; l < nrows; ++l) {
      cur = rb[l * TAGS + cur];        // LDS broadcast; all lanes track cur
      if (l == lane) mytag = cur;      // lane l owns step (cbase - l)
    }
    if (lane < nrows) out[(size_t)b * SEQ + (cbase - lane)] = (float)mytag;
    buf ^= 1;
  }
}

// ---------------------------------------------------------------------------
extern "C" void kernel_launch(void* const* d_in, const int* in_sizes, int n_in,
                              void* d_out, int out_size, void* d_ws, size_t ws_size,
                              hipStream_t stream) {
  (void)in_sizes; (void)n_in; (void)out_size; (void)ws_size;
  const float* em     = (const float*)d_in[0];  // emissions [512,1024,48] f32
  // d_in[1] = mask (int32) -- unused by reference
  const float* trans  = (const float*)d_in[2];  // [48,48]
  const float* startT = (const float*)d_in[3];  // [48]
  const float* endT   = (const float*)d_in[4];  // [48]
  float* out = (float*)d_out;                   // [512,1024]

  // Workspace layout: u8 backpointers [1023,512,48] (25,141,248 B), then
  // int finalTag[512].
  uint8_t* bp = (uint8_t*)d_ws;
  int* finalTag = (int*)((char*)d_ws + (size_t)(SEQ - 1) * BATCH * TAGS);

  dim3 blk(WPB * 32);
  dim3 grd(BATCH / WPB);
  viterbi_forward<<<grd, blk, 0, stream>>>(em, trans, startT, endT, bp, finalTag);
  viterbi_backtrace<<<grd, blk, 0, stream>>>(bp, finalTag, out);
}